// GNNModel_78194174591377
// MI455X (gfx1250) — compile-verified
//
#include <hip/hip_runtime.h>
#include <math.h>

#define N_NODES   50000
#define N_EDGES   800000
#define IN_DIM    768
#define HID_DIM   128
#define OUT_DIM   64
#define N_CLASSES 2

typedef __attribute__((ext_vector_type(2))) float v2f;
typedef __attribute__((ext_vector_type(8))) float v8f;

// ---------------- degree / normalization ----------------

__global__ void k_deg_init(float* __restrict__ deg, int n) {
    int i = blockIdx.x * blockDim.x + threadIdx.x;
    if (i < n) deg[i] = 1.0f;               // self-loop contribution
}

__global__ void k_deg_count(const int* __restrict__ dst, float* __restrict__ deg, int e) {
    int i = blockIdx.x * blockDim.x + threadIdx.x;
    if (i < e) atomicAdd(&deg[dst[i]], 1.0f);
}

__global__ void k_rsqrt(float* __restrict__ deg, int n) {
    int i = blockIdx.x * blockDim.x + threadIdx.x;
    if (i < n) deg[i] = rsqrtf(deg[i]);     // dis = deg^{-1/2}, in place
}

// ---------------- WMMA f32 GEMM: C[M x N] = A[M x K] @ B[K x N] ----------------
// One wave computes one 16x16 output tile using V_WMMA_F32_16X16X4_F32.
// A layout (16x4 f32): lane<16 -> M=lane, holds K=k,k+1 ; lane>=16 -> M=lane-16, K=k+2,k+3
// B layout (4x16 f32): v0 = rows {k (lanes0-15), k+2 (lanes16-31)}, v1 = rows {k+1, k+3}
// C/D layout: VGPR i -> row (i + 8*(lane>=16)), col = lane&15

template<int KDIM, int NDIM, bool RELU_A>
__global__ __launch_bounds__(256)
void k_gemm_wmma(const float* __restrict__ A, const float* __restrict__ B,
                 float* __restrict__ C, int mtiles) {
    const int ntiles = NDIM / 16;
    int wave = (blockIdx.x * blockDim.x + threadIdx.x) >> 5;
    int lane = threadIdx.x & 31;
    if (wave >= mtiles * ntiles) return;    // wave-uniform: EXEC stays all-ones
    int mt = wave / ntiles;
    int nt = wave % ntiles;

    int m  = lane & 15;                     // also the column index n for B
    int kh = (lane >> 4) << 1;              // 0 or 2

    const float* arow = A + (mt * 16 + m) * KDIM + kh;
    const float* bcol = B + nt * 16 + m;

    v8f acc = {};
    #pragma unroll 4
    for (int k = 0; k < KDIM; k += 4) {
        float2 av = *(const float2*)(arow + k);
        v2f a, b;
        a.x = av.x; a.y = av.y;
        if (RELU_A) { a.x = fmaxf(a.x, 0.0f); a.y = fmaxf(a.y, 0.0f); }
        b.x = bcol[(k + kh)     * NDIM];
        b.y = bcol[(k + kh + 1) * NDIM];
        acc = __builtin_amdgcn_wmma_f32_16x16x4_f32(
                  false, a, false, b, (short)0, acc, false, false);
    }

    int row0 = mt * 16 + ((lane >> 4) << 3);
    int col  = nt * 16 + (lane & 15);
    #pragma unroll
    for (int i = 0; i < 8; ++i)
        C[(row0 + i) * NDIM + col] = acc[i];
}

// ---------------- self-loop + bias init: acc = dis^2 * xw + b ----------------

template<int F>
__global__ void k_self_init(const float* __restrict__ xw, const float* __restrict__ dis,
                            const float* __restrict__ bias, float* __restrict__ acc) {
    int t = blockIdx.x * blockDim.x + threadIdx.x;  // one float4
    const int Q = F / 4;
    if (t >= N_NODES * Q) return;
    int node = t / Q;
    int f    = (t % Q) * 4;
    float s = dis[node]; s = s * s;
    float4 v  = *(const float4*)(xw   + node * F + f);
    float4 bb = *(const float4*)(bias + f);
    float4 r;
    r.x = s * v.x + bb.x;
    r.y = s * v.y + bb.y;
    r.z = s * v.z + bb.z;
    r.w = s * v.w + bb.w;
    *(float4*)(acc + node * F + f) = r;
}

// ---------------- edge scatter: acc[dst] += dis[src]*dis[dst] * xw[src] ----------------
// Consecutive lanes cover consecutive float4 features of one edge (F=128 -> 1 wave/edge).

template<int F>
__global__ void k_scatter(const int* __restrict__ src, const int* __restrict__ dst,
                          const float* __restrict__ dis, const float* __restrict__ xw,
                          float* __restrict__ acc) {
    const int Q = F / 4;
    int t = blockIdx.x * blockDim.x + threadIdx.x;
    int e = t / Q;
    if (e >= N_EDGES) return;
    int f = (t % Q) * 4;
    int s = src[e], d = dst[e];
    float c = dis[s] * dis[d];
    float4 v = *(const float4*)(xw + s * F + f);
    float* p = acc + d * F + f;
    atomicAdd(p + 0, c * v.x);
    atomicAdd(p + 1, c * v.y);
    atomicAdd(p + 2, c * v.z);
    atomicAdd(p + 3, c * v.w);
}

// ---------------- classifier: out = h2 @ Wc + bc (N x 64 -> N x 2) ----------------

__global__ void k_out(const float* __restrict__ h, const float* __restrict__ Wc,
                      const float* __restrict__ bc, float* __restrict__ out) {
    int n = blockIdx.x * blockDim.x + threadIdx.x;
    if (n >= N_NODES) return;
    float o0 = bc[0], o1 = bc[1];
    const float* row = h + n * OUT_DIM;
    #pragma unroll 8
    for (int f = 0; f < OUT_DIM; ++f) {
        float v = row[f];
        o0 += v * Wc[f * 2 + 0];
        o1 += v * Wc[f * 2 + 1];
    }
    out[n * 2 + 0] = o0;
    out[n * 2 + 1] = o1;
}

// ---------------- launcher ----------------

extern "C" void kernel_launch(void* const* d_in, const int* in_sizes, int n_in,
                              void* d_out, int out_size, void* d_ws, size_t ws_size,
                              hipStream_t stream) {
    const float* x  = (const float*)d_in[0];
    const int*   ei = (const int*)d_in[1];     // [2, E]: row 0 = src, row 1 = dst
    const float* W1 = (const float*)d_in[2];
    const float* b1 = (const float*)d_in[3];
    const float* W2 = (const float*)d_in[4];
    const float* b2 = (const float*)d_in[5];
    const float* Wc = (const float*)d_in[6];
    const float* bc = (const float*)d_in[7];
    float* out = (float*)d_out;

    const int* srcIdx = ei;
    const int* dstIdx = ei + N_EDGES;

    // workspace layout (floats), deterministic aliasing across the stream-ordered sequence
    float* wsf  = (float*)d_ws;
    float* dis  = wsf;                      // 50000 (padded to 50048)
    float* bufA = wsf + 50048;              // xw1 [N,128], later hw2 [N,64]
    float* bufB = bufA + N_NODES * HID_DIM; // acc1 [N,128], later acc2 [N,64]

    const int B = 256;
    const int MT = N_NODES / 16;            // 3125 exactly

    // degree / dis
    k_deg_init <<<(N_NODES + B - 1) / B, B, 0, stream>>>(dis, N_NODES);
    k_deg_count<<<(N_EDGES + B - 1) / B, B, 0, stream>>>(dstIdx, dis, N_EDGES);
    k_rsqrt    <<<(N_NODES + B - 1) / B, B, 0, stream>>>(dis, N_NODES);

    // layer 1: xw1 = x @ W1
    {
        int waves  = MT * (HID_DIM / 16);   // 25000
        int blocks = (waves + 7) / 8;       // 3125
        k_gemm_wmma<IN_DIM, HID_DIM, false><<<blocks, B, 0, stream>>>(x, W1, bufA, MT);
    }
    k_self_init<HID_DIM><<<(N_NODES * HID_DIM / 4 + B - 1) / B, B, 0, stream>>>(bufA, dis, b1, bufB);
    k_scatter<HID_DIM><<<(N_EDGES * (HID_DIM / 4) + B - 1) / B, B, 0, stream>>>(srcIdx, dstIdx, dis, bufA, bufB);

    // layer 2: hw2 = relu(acc1) @ W2   (ReLU fused into A load)
    {
        int waves  = MT * (OUT_DIM / 16);   // 12500
        int blocks = (waves + 7) / 8;       // 1563
        k_gemm_wmma<HID_DIM, OUT_DIM, true><<<blocks, B, 0, stream>>>(bufB, W2, bufA, MT);
    }
    k_self_init<OUT_DIM><<<(N_NODES * OUT_DIM / 4 + B - 1) / B, B, 0, stream>>>(bufA, dis, b2, bufB);
    k_scatter<OUT_DIM><<<(N_EDGES * (OUT_DIM / 4) + B - 1) / B, B, 0, stream>>>(srcIdx, dstIdx, dis, bufA, bufB);

    // classifier
    k_out<<<(N_NODES + B - 1) / B, B, 0, stream>>>(bufB, Wc, bc, out);

    (void)in_sizes; (void)n_in; (void)out_size; (void)ws_size;
}